// Model_60026462929565
// MI455X (gfx1250) — compile-verified
//
#include <hip/hip_runtime.h>
#include <hip/hip_bf16.h>
#include <stdint.h>

// Problem constants (match reference)
#define DD      128
#define NREL    8
#define NCHUNK  9            // 8 relations + self
#define NLAYERS 4
#define PSTR    288          // floats per K-pair row in packed layout (256+32 pad)
#define TILE_DW (64 * PSTR)  // dwords per packed weight tile (18432)

typedef __attribute__((ext_vector_type(2))) float        v2f;
typedef __attribute__((ext_vector_type(8))) float        v8f;
typedef __attribute__((ext_vector_type(4))) unsigned int u32x4;
typedef __attribute__((ext_vector_type(8))) int          i32x8;
typedef __attribute__((ext_vector_type(4))) int          i32x4;

// ---------------------------------------------------------------------------
// Zero the per-relation aggregation buffer (R*N*D floats) with float4 stores.
// ---------------------------------------------------------------------------
__global__ void zero_f4(float4* __restrict__ p, long long n4) {
  long long i = (long long)blockIdx.x * blockDim.x + threadIdx.x;
  if (i < n4) p[i] = make_float4(0.f, 0.f, 0.f, 0.f);
}

// ---------------------------------------------------------------------------
// One-shot weight repack into the LDS image layout:
//   wpack[l][chunk][k/2][n*2 + (k&1)]  (pair stride PSTR floats, 32-fl pad)
// Pad region keeps whatever ws holds; it is copied to LDS but never read.
// chunk 0..7 = W_rel[l][chunk], chunk 8 = W_self[l].
// ---------------------------------------------------------------------------
__global__ void __launch_bounds__(256)
repack_weights(const float* __restrict__ Wrel,   // [L][R][DD][DD]
               const float* __restrict__ Wself,  // [L][DD][DD]
               float* __restrict__ wpack) {      // [L][NCHUNK][64][PSTR]
  long long idx = (long long)blockIdx.x * blockDim.x + threadIdx.x;
  const long long total = (long long)NLAYERS * NCHUNK * DD * DD;
  if (idx >= total) return;
  const int n     = (int)(idx & (DD - 1));
  const int k     = (int)((idx >> 7) & (DD - 1));
  const int chunk = (int)((idx >> 14) % NCHUNK);
  const int l     = (int)(idx / ((long long)NCHUNK << 14));
  const float* srcM = (chunk < NREL)
                          ? (Wrel + ((size_t)l * NREL + chunk) * DD * DD)
                          : (Wself + (size_t)l * DD * DD);
  const float v = srcM[k * DD + n];
  wpack[(((size_t)l * NCHUNK + chunk) * 64 + (k >> 1)) * PSTR + n * 2 + (k & 1)] = v;
}

// ---------------------------------------------------------------------------
// Edge scatter: agg[et[e]][dst[e]][:] += h[src[e]][:]
// One wave32 per edge; each lane handles 4 consecutive floats (128 = 32*4).
// Uses native global_atomic_add_f32.
// ---------------------------------------------------------------------------
__global__ void __launch_bounds__(256)
scatter_edges(const float* __restrict__ hin,
              const int*   __restrict__ src,
              const int*   __restrict__ dst,
              const int*   __restrict__ et,
              float*       __restrict__ agg,
              int nEdges, int nNodes) {
  const int e    = (int)((blockIdx.x * blockDim.x + threadIdx.x) >> 5);
  const int lane = threadIdx.x & 31;
  if (e >= nEdges) return;
  const int s = src[e];
  const int d = dst[e];
  const int r = et[e];
  const float4 v = ((const float4*)(hin + (size_t)s * DD))[lane];
  float* o = agg + ((size_t)r * nNodes + d) * DD + lane * 4;
#if defined(__HIP_DEVICE_COMPILE__)
  unsafeAtomicAdd(o + 0, v.x);
  unsafeAtomicAdd(o + 1, v.y);
  unsafeAtomicAdd(o + 2, v.z);
  unsafeAtomicAdd(o + 3, v.w);
#else
  atomicAdd(o + 0, v.x); atomicAdd(o + 1, v.y);
  atomicAdd(o + 2, v.z); atomicAdd(o + 3, v.w);
#endif
}

// ---------------------------------------------------------------------------
// Issue (no wait) a TDM copy of one packed weight tile (TILE_DW dwords, dense
// 1-D) from global into LDS.  D# per CDNA5 ISA §8.3/8.4, data_size=4B.
// EXEC is ignored by TENSOR_LOAD_TO_LDS; issued by one wave only.
// 6-arg builtin: (u32x4 g0, i32x8 g1, i32x4 g2, i32x4 g3, i32x8 g4, i32 cpol)
// ---------------------------------------------------------------------------
__device__ __forceinline__ void tdm_issue_tile(const float* __restrict__ B,
                                               float* __restrict__ sBuf) {
#if __has_builtin(__builtin_amdgcn_tensor_load_to_lds)
  // LDS aperture addresses carry the LDS byte offset in bits [31:0].
  const unsigned lds_off = (unsigned)(uintptr_t)(&sBuf[0]);
  const unsigned long long ga = (unsigned long long)(uintptr_t)B;
  // Group0: count=1, lds_addr, global_addr[56:0], type=2 ("image")
  u32x4 g0 = { 1u,
               lds_off,
               (unsigned)(ga & 0xffffffffu),
               (unsigned)((ga >> 32) & 0x1ffffffu) | (2u << 30) };
  // Group1: wg_mask=0, data_size=2 (4B); dense 1-D copy of TILE_DW dwords:
  // tensor_dim0 = tile_dim0 = TILE_DW, tensor_dim1 = tile_dim1 = 1,
  // tensor_dim0_stride = TILE_DW.
  i32x8 g1 = { (int)(2u << 16),          // data_size=4B
               (int)(TILE_DW << 16),     // tensor_dim0 low16
               (int)(1 << 16),           // tensor_dim0 hi=0 | tensor_dim1 lo16
               (int)(TILE_DW << 16),     // tensor_dim1 hi=0 | tile_dim0
               1,                        // tile_dim1=1 (tile_dim2=0)
               TILE_DW,                  // tensor_dim0_stride low32
               0, 0 };
  i32x4 gz4 = { 0, 0, 0, 0 };
  i32x8 gz8 = { 0, 0, 0, 0, 0, 0, 0, 0 };
  __builtin_amdgcn_tensor_load_to_lds(g0, g1, gz4, gz4, gz8, 0);
#else
  for (int i = threadIdx.x & 31; i < TILE_DW / 4; i += 32)
    ((float4*)sBuf)[i] = ((const float4*)B)[i];
#endif
}

// ---------------------------------------------------------------------------
// Fused layer GEMM:
//   hout = relu( bias + hin @ Wself + sum_r agg_r @ Wrel[r] )
// Block = 256 threads (8 wave32s), 128 output rows per block.
// Per wave: 16x128 output strip via V_WMMA_F32_16X16X4_F32.
//   C (16x16 f32): 8 VGPRs; lanes 0-15 -> M=v, lanes 16-31 -> M=v+8, N=lane&15
//   A (16x4  f32): lane&15 = M, VGPR pair = K = 2*(lane>>4) + {0,1}
//   B (4x16  f32): lane&15 = N, VGPR pair = K = 2*(lane>>4) + {0,1}
// Weight tiles (pre-packed K-pair-interleaved) double-buffered in LDS and
// streamed by the TDM one chunk ahead (s_wait_tensorcnt 1 = current tile
// resident).  B fragment = single contiguous ds_load_b64, conflict-free
// (even pairs -> banks 0-31, odd pairs -> banks 32-63 via 32-float pad).
// A strip register-preloaded per chunk (one batched loadcnt window).
// ---------------------------------------------------------------------------
__global__ void __launch_bounds__(256)
rgcn_layer_gemm(const float* __restrict__ hin,
                const float* __restrict__ agg,     // [NREL][nNodes][DD]
                const float* __restrict__ wpackL,  // [NCHUNK][64][PSTR]
                const float* __restrict__ bias,    // [DD]
                float*       __restrict__ hout,
                int nNodes) {
  __shared__ float sB[2][TILE_DW];               // 2 x 72 KB packed tiles

  const int lane    = threadIdx.x & 31;
  const int wave    = threadIdx.x >> 5;
  const int n16     = lane & 15;                 // N (and A-row M) index
  const int khalf   = lane >> 4;                 // selects K pair {0,1} vs {2,3}
  const int rowBase = blockIdx.x * 128 + wave * 16;

  v8f c[8] = {};                                 // 16x128 f32 accumulators

  // Prologue: start streaming the first packed tile.
  if (threadIdx.x < 32) tdm_issue_tile(wpackL, &sB[0][0]);

  int row = rowBase + n16;
  if (row >= nNodes) row = nNodes - 1;           // clamp (stores are guarded)

  for (int chunk = 0; chunk < NCHUNK; ++chunk) {
    // All waves are done reading sB[(chunk+1)&1] (last used in chunk-1).
    __syncthreads();
    if (threadIdx.x < 32) {
      if (chunk + 1 < NCHUNK) {
        tdm_issue_tile(wpackL + (size_t)(chunk + 1) * TILE_DW,
                       &sB[(chunk + 1) & 1][0]);
        __builtin_amdgcn_s_wait_tensorcnt(1);    // tile for 'chunk' resident
      } else {
        __builtin_amdgcn_s_wait_tensorcnt(0);    // final tile resident
      }
    }
    __syncthreads();                             // tile visible to all waves

    const float* sBuf = &sB[chunk & 1][0];
    const float* A = (chunk < NREL) ? (agg + (size_t)chunk * nNodes * DD) : hin;
    const float* Arow = A + (size_t)row * DD + 2 * khalf;

    // Register-preload this wave's 16x128 A strip (per-lane: 32 x float2).
    v2f aReg[32];
#pragma unroll
    for (int t = 0; t < 32; ++t)
      aReg[t] = *(const v2f*)(Arow + 4 * t);

#pragma unroll
    for (int t = 0; t < 32; ++t) {
      // K pair index for this lane: (4t + 2*khalf)/2 = 2t + khalf.
      const float* bRow = sBuf + (size_t)(2 * t + khalf) * PSTR + n16 * 2;
#pragma unroll
      for (int j = 0; j < 8; ++j) {
        const v2f bF = *(const v2f*)(bRow + j * 32);   // one ds_load_b64
#if __has_builtin(__builtin_amdgcn_wmma_f32_16x16x4_f32)
        c[j] = __builtin_amdgcn_wmma_f32_16x16x4_f32(
            /*neg_a=*/false, aReg[t], /*neg_b=*/false, bF,
            /*c_mod=*/(short)0, c[j], /*reuse_a=*/false, /*reuse_b=*/false);
#else
        c[j][0] += aReg[t].x * bF.x + aReg[t].y * bF.y;
#endif
      }
    }
  }

  // Epilogue: bias + ReLU + store, honoring the 16x16 f32 C layout.
  const int mBase = rowBase + 8 * khalf;
#pragma unroll
  for (int j = 0; j < 8; ++j) {
    const float bv = bias[j * 16 + n16];
#pragma unroll
    for (int v = 0; v < 8; ++v) {
      const int r2 = mBase + v;
      if (r2 < nNodes) {
        float x = c[j][v] + bv;
        hout[(size_t)r2 * DD + j * 16 + n16] = x > 0.f ? x : 0.f;
      }
    }
  }
}

// ---------------------------------------------------------------------------
// Host-side orchestration (graph-capture safe: only kernel launches).
// Workspace layout: agg [R*N*D] | h0 [N*D] | h1 [N*D] | wpack [L*9*TILE_DW]
// Once:     repack weights into the LDS-image layout.
// Layer l:  zero agg -> scatter raw features per relation -> fused GEMM.
// ---------------------------------------------------------------------------
extern "C" void kernel_launch(void* const* d_in, const int* in_sizes, int n_in,
                              void* d_out, int out_size, void* d_ws, size_t ws_size,
                              hipStream_t stream) {
  const float* feat  = (const float*)d_in[0];
  const float* Wrel  = (const float*)d_in[1];  // [NLAYERS][NREL][DD][DD]
  const float* Wself = (const float*)d_in[2];  // [NLAYERS][DD][DD]
  const float* bias  = (const float*)d_in[3];  // [NLAYERS][DD]
  const int*   src   = (const int*)d_in[4];
  const int*   dst   = (const int*)d_in[5];
  const int*   et    = (const int*)d_in[6];

  const int nNodes = in_sizes[0] / DD;
  const int nEdges = in_sizes[4];

  float* agg   = (float*)d_ws;
  float* h0    = agg + (size_t)NREL * nNodes * DD;
  float* h1    = h0 + (size_t)nNodes * DD;
  float* wpack = h1 + (size_t)nNodes * DD;

  const long long n4      = (long long)NREL * nNodes * DD / 4;
  const int zeroBlocks    = (int)((n4 + 255) / 256);
  const int scatterBlocks = (nEdges + 7) / 8;           // 8 waves / block
  const int gemmBlocks    = (nNodes + 127) / 128;
  const long long nRepack = (long long)NLAYERS * NCHUNK * DD * DD;
  const int repackBlocks  = (int)((nRepack + 255) / 256);

  repack_weights<<<repackBlocks, 256, 0, stream>>>(Wrel, Wself, wpack);

  for (int l = 0; l < NLAYERS; ++l) {
    const float* hin  = (l == 0) ? feat : ((l & 1) ? h0 : h1);
    float*       hout = (l == NLAYERS - 1) ? (float*)d_out
                                           : ((l & 1) ? h1 : h0);

    zero_f4<<<zeroBlocks, 256, 0, stream>>>((float4*)agg, n4);
    scatter_edges<<<scatterBlocks, 256, 0, stream>>>(hin, src, dst, et, agg,
                                                     nEdges, nNodes);
    rgcn_layer_gemm<<<gemmBlocks, 256, 0, stream>>>(
        hin, agg,
        wpack + (size_t)l * NCHUNK * TILE_DW,
        bias + (size_t)l * DD,
        hout, nNodes);
  }
}